// HyperRingOperator_76441827934261
// MI455X (gfx1250) — compile-verified
//
#include <hip/hip_runtime.h>
#include <math.h>
#include <stdint.h>

// HyperRingOperator on MI455X (gfx1250).
// Only columns 0..1 of the [B, 2048] inputs are live. Gather the strided
// (stride = 8KB) 8-byte column pairs into LDS with the Tensor Data Mover,
// then evaluate the 32-point closed-loop line integral per row.

#define NUM_POINTS     32
#define ROWS_PER_BLOCK 128
#define D_FIXED        2048

typedef unsigned int u32x4 __attribute__((ext_vector_type(4)));
typedef int          i32x4 __attribute__((ext_vector_type(4)));
typedef int          i32x8 __attribute__((ext_vector_type(8)));

// Issue one TDM load of a [rows x 2] fp32 tile (row stride ld_elems) into LDS.
// D# group0/group1 packed per CDNA5 ISA §8.3/§8.4; groups 2/3 zero (2D tensor).
__device__ __forceinline__ void tdm_load_tile_2col(unsigned lds_byte_off,
                                                   unsigned long long gaddr,
                                                   unsigned rows,
                                                   unsigned ld_elems) {
  u32x4 g0;
  g0.x = 1u;                                      // count=1, is_restore=0, gather off
  g0.y = lds_byte_off;                            // lds_addr (bytes)
  g0.z = (unsigned)(gaddr & 0xFFFFFFFFull);       // global_addr[31:0]
  g0.w = (unsigned)((gaddr >> 32) & 0x01FFFFFFull)// global_addr[56:32]
         | (2u << 30);                            // type = 2 ("image")

  i32x8 g1;
  g1[0] = (int)(2u << 16);                        // data_size = 2 -> 4 bytes
  g1[1] = (int)((ld_elems & 0xFFFFu) << 16);      // tensor_dim0[15:0] in [31:16]
  g1[2] = (int)(((ld_elems >> 16) & 0xFFFFu)      // tensor_dim0[31:16]
         | ((rows & 0xFFFFu) << 16));             // tensor_dim1[15:0]
  g1[3] = (int)(((rows >> 16) & 0xFFFFu)          // tensor_dim1[31:16]
         | (2u << 16));                           // tile_dim0 = 2 elements (8B)
  g1[4] = (int)(rows & 0xFFFFu);                  // tile_dim1 = rows, tile_dim2 = 0
  g1[5] = (int)ld_elems;                          // tensor_dim0_stride[31:0]
  g1[6] = 0;                                      // stride hi / dim1_stride lo
  g1[7] = 0;                                      // dim1_stride hi

  i32x4 z4 = {0, 0, 0, 0};
#if __clang_major__ >= 23
  i32x8 z8 = {0, 0, 0, 0, 0, 0, 0, 0};
  __builtin_amdgcn_tensor_load_to_lds(g0, g1, z4, z4, z8, 0);
#else
  __builtin_amdgcn_tensor_load_to_lds(g0, g1, z4, z4, 0);
#endif
}

__global__ void __launch_bounds__(ROWS_PER_BLOCK)
hyper_ring_kernel(const float* __restrict__ residue,
                  const float* __restrict__ manifold,
                  float* __restrict__ out, int B, int D) {
  __shared__ float sR[2 * ROWS_PER_BLOCK];   // residue  cols 0..1 per row
  __shared__ float sM[2 * ROWS_PER_BLOCK];   // manifold cols 0..1 per row
  __shared__ float cTab[NUM_POINTS];
  __shared__ float sTab[NUM_POINTS];

  const int row0 = blockIdx.x * ROWS_PER_BLOCK;
  int rows = B - row0;
  if (rows > ROWS_PER_BLOCK) rows = ROWS_PER_BLOCK;
  const int t = threadIdx.x;

  if (t < 32) {
    // cos/sin table for angles = linspace(0, 2*pi, 32).
    // angle_i = 2*pi * i/31, i.e. already < one full revolution, so the
    // hardware V_SIN_F32/V_COS_F32 fast path (__sinf/__cosf = one v_mul by
    // 1/(2*pi) + one trans op) needs no range reduction and its ~2^-21
    // error is far below this output's FP-noise floor.
    const float step = 6.28318530717958647692f / (float)(NUM_POINTS - 1);
    float a = (float)t * step;
    cTab[t] = __cosf(a);
    sTab[t] = __sinf(a);

    // One TDM issue per (first) wave: descriptors are wave-uniform
    // (blockIdx + kernel args). EXEC-masked lanes don't matter; the tensor
    // op issues once per wave and is tracked with TENSORcnt.
    tdm_load_tile_2col((unsigned)(uintptr_t)&sR[0],
                       (unsigned long long)(uintptr_t)(residue + (size_t)row0 * (size_t)D),
                       (unsigned)rows, (unsigned)D);
    tdm_load_tile_2col((unsigned)(uintptr_t)&sM[0],
                       (unsigned long long)(uintptr_t)(manifold + (size_t)row0 * (size_t)D),
                       (unsigned)rows, (unsigned)D);
    __builtin_amdgcn_s_wait_tensorcnt(0);     // DMA data landed in LDS
  }
  __syncthreads();
  asm volatile("" ::: "memory");              // LDS was written behind the compiler's back

  if (t < rows) {
    float r0 = sR[2 * t + 0], r1 = sR[2 * t + 1];
    float m0 = sM[2 * t + 0], m1 = sM[2 * t + 1];

    float g0 = r0 - m0;
    float g1 = r1 - m1;
    float radius = sqrtf(g0 * g0 + g1 * g1) + 1e-6f;

    // Closed-loop line integral, faithful to the reference op order:
    // build points, forward diffs, wrap segment, mean.
    float x0 = m0 + radius * cTab[0];
    float y0 = m1 + radius * sTab[0];
    float xp = x0, yp = y0, acc = 0.0f;
#pragma unroll
    for (int i = 1; i < NUM_POINTS; ++i) {
      float xi = m0 + radius * cTab[i];
      float yi = m1 + radius * sTab[i];
      acc += g0 * (xi - xp) + g1 * (yi - yp);
      xp = xi;
      yp = yi;
    }
    acc += g0 * (x0 - xp) + g1 * (y0 - yp);    // wrap segment

    out[row0 + t] = acc * (1.0f / (float)NUM_POINTS);
  }
}

extern "C" void kernel_launch(void* const* d_in, const int* in_sizes, int n_in,
                              void* d_out, int out_size, void* d_ws, size_t ws_size,
                              hipStream_t stream) {
  (void)n_in; (void)d_ws; (void)ws_size;
  const float* residue  = (const float*)d_in[0];
  const float* manifold = (const float*)d_in[1];
  float* out = (float*)d_out;

  const int D = D_FIXED;                       // feature dim per reference
  const int B = in_sizes[0] / D;               // 32768
  (void)out_size;

  const int grid = (B + ROWS_PER_BLOCK - 1) / ROWS_PER_BLOCK;
  hyper_ring_kernel<<<grid, ROWS_PER_BLOCK, 0, stream>>>(residue, manifold, out, B, D);
}